// ForwardDeformerDiffusedSkinning_89687507075414
// MI455X (gfx1250) — compile-verified
//
#include <hip/hip_runtime.h>
#include <hip/hip_bf16.h>
#include <math.h>

typedef __attribute__((ext_vector_type(2))) float v2f;
typedef __attribute__((ext_vector_type(8))) float v8f;

#define RGRID   128
#define NCH     24
#define NVOX    (RGRID*RGRID*RGRID)      // 2097152
#define NPTS    20000
#define NINIT   9
#define NSOLVE  (NPTS*NINIT)             // 180000
#define MAXSTEPS 50
#define CVG_T   1e-5f
#define DVG_T   1.0f
#define EPS_B   1e-6f
#define WPB     6                        // waves per block
#define THREADS (WPB*32)                 // 192
#define SOLVE_BLOCKS (NSOLVE/(WPB*16))   // 1875 (exact)

// ---------------------------------------------------------------------------
// Setup: flatten bone transforms T[24][12] (rows 0..2 of each 4x4) and the
// affine inverses of the 9 INIT_BONES transforms (last row is [0,0,0,1]).
// ---------------------------------------------------------------------------
__global__ void deformer_setup_kernel(const float* __restrict__ tfs,
                                      float* __restrict__ Tmat,
                                      float* __restrict__ invT) {
  if (threadIdx.x == 0 && blockIdx.x == 0) {
    for (int n = 0; n < NCH; ++n)
      for (int e = 0; e < 12; ++e)
        Tmat[n*12 + e] = tfs[n*16 + e];
    const int ib[NINIT] = {0,1,2,4,5,16,17,18,19};
    for (int b = 0; b < NINIT; ++b) {
      const float* M = tfs + ib[b]*16;
      float a00=M[0],a01=M[1],a02=M[2],t0=M[3];
      float a10=M[4],a11=M[5],a12=M[6],t1=M[7];
      float a20=M[8],a21=M[9],a22=M[10],t2=M[11];
      float det = a00*(a11*a22-a12*a21) - a01*(a10*a22-a12*a20) + a02*(a10*a21-a11*a20);
      float id  = 1.0f/det;
      float r00=(a11*a22-a12*a21)*id, r01=(a02*a21-a01*a22)*id, r02=(a01*a12-a02*a11)*id;
      float r10=(a12*a20-a10*a22)*id, r11=(a00*a22-a02*a20)*id, r12=(a02*a10-a00*a12)*id;
      float r20=(a10*a21-a11*a20)*id, r21=(a01*a20-a00*a21)*id, r22=(a00*a11-a01*a10)*id;
      float* o = invT + b*12;
      o[0]=r00; o[1]=r01; o[2]=r02;  o[3]=-(r00*t0+r01*t1+r02*t2);
      o[4]=r10; o[5]=r11; o[6]=r12;  o[7]=-(r10*t0+r11*t1+r12*t2);
      o[8]=r20; o[9]=r21; o[10]=r22; o[11]=-(r20*t0+r21*t1+r22*t2);
    }
  }
}

// ---------------------------------------------------------------------------
// Transpose weight grid [C,D,H,W] -> [D,H,W,C] (voxel-major) through LDS so
// both global read and write phases are coalesced. One-time 192MB -> L2.
// ---------------------------------------------------------------------------
__global__ __launch_bounds__(256) void grid_transpose_kernel(
    const float* __restrict__ g, float* __restrict__ gt) {
  __shared__ float t[NCH][257];
  const int tid  = threadIdx.x;
  const int vox0 = blockIdx.x * 256;
#pragma unroll
  for (int c = 0; c < NCH; ++c)
    t[c][tid] = g[(size_t)c*NVOX + vox0 + tid];
  __syncthreads();
  const size_t outBase = (size_t)vox0 * NCH;
#pragma unroll
  for (int k = 0; k < NCH; ++k) {
    int f  = k*256 + tid;
    int vv = f / NCH;
    int c  = f - vv*NCH;
    gt[outBase + f] = t[c][vv];
  }
}

// ---------------------------------------------------------------------------
// Main solver. TR=true: voxel-major transposed grid (float4 gathers).
// Wave32 handles 16 solves; lane pair (m, m+16) redundantly owns solve m and
// splits the 24 channels 12/12 for sampling. Weight->blend GEMM
// W[16x24] @ T[24x12] done with 6x v_wmma_f32_16x16x4_f32 per iteration.
// ---------------------------------------------------------------------------
template<bool TR>
__global__ __launch_bounds__(THREADS) void broyden_solve_kernel(
    const float* __restrict__ xd,   const float* __restrict__ grid,
    const float* __restrict__ Tmat, const float* __restrict__ invT,
    const float* __restrict__ pext, const float* __restrict__ pcen,
    float* __restrict__ out) {
  __shared__ float    lds[WPB*576];     // per wave: W[16][24] + D[16][12]
  __shared__ unsigned s_vote[WPB];

  const int tid  = threadIdx.x;
  const int wv   = tid >> 5;
  const int lane = tid & 31;
  const int m    = lane & 15;           // solve slot / WMMA row
  const int h    = lane >> 4;           // channel half / K half
  float* ldsW = lds + wv*576;
  float* ldsD = ldsW + 384;

  int s = ((int)blockIdx.x*WPB + wv)*16 + m;
  if (s >= NSOLVE) s = NSOLVE - 1;
  const int p  = s / NINIT;
  const int bi = s - p*NINIT;

  const float ext  = pext[0];
  const float cen0 = pcen[0], cen1 = pcen[1], cen2 = pcen[2];
  const float sclv = 2.0f / ext;

  const float xd0 = xd[3*p+0], xd1 = xd[3*p+1], xd2 = xd[3*p+2];

  const float* IT = invT + bi*12;
  float x0 = IT[0]*xd0 + IT[1]*xd1 + IT[2]*xd2  + IT[3];
  float x1 = IT[4]*xd0 + IT[5]*xd1 + IT[6]*xd2  + IT[7];
  float x2 = IT[8]*xd0 + IT[9]*xd1 + IT[10]*xd2 + IT[11];

  const int c0   = h*12;   // first channel this lane samples
  const int ncol = m;      // WMMA N column owned by this lane

  // B fragments for v_wmma_f32_16x16x4_f32 (K=24 -> 6 k-tiles), loop invariant.
  // VGPR0: lanes0-15 K=k0, lanes16-31 K=k0+2 ; VGPR1: K=k0+1 / K=k0+3.
  v2f Bf[6];
#pragma unroll
  for (int kt = 0; kt < 6; ++kt) {
    const int k0 = 4*kt + 2*h;
    float b0 = 0.f, b1 = 0.f;
    if (ncol < 12) { b0 = Tmat[k0*12 + ncol]; b1 = Tmat[(k0+1)*12 + ncol]; }
    Bf[kt][0] = b0; Bf[kt][1] = b1;
  }

  // ------- initial evaluation: gx = g(x0), analytic Jacobian, J_inv0 -------
  float gx0, gx1, gx2;
  float Ji00,Ji01,Ji02,Ji10,Ji11,Ji12,Ji20,Ji21,Ji22;
  {
    float vx=(x0-cen0)*sclv, vy=(x1-cen1)*sclv, vz=(x2-cen2)*sclv;
    float pxu=((vx+1.f)*(float)RGRID-1.f)*0.5f;
    float pyu=((vy+1.f)*(float)RGRID-1.f)*0.5f;
    float pzu=((vz+1.f)*(float)RGRID-1.f)*0.5f;
    float px=fminf(fmaxf(pxu,0.f),127.f), py=fminf(fmaxf(pyu,0.f),127.f), pz=fminf(fmaxf(pzu,0.f),127.f);
    int ix0=(int)floorf(px), iy0=(int)floorf(py), iz0=(int)floorf(pz);
    float wxf=px-(float)ix0, wyf=py-(float)iy0, wzf=pz-(float)iz0;
    int xi[2]={ix0,min(ix0+1,RGRID-1)}, yi[2]={iy0,min(iy0+1,RGRID-1)}, zi[2]={iz0,min(iz0+1,RGRID-1)};
    float fx[2]={1.f-wxf,wxf}, fy[2]={1.f-wyf,wyf}, fz[2]={1.f-wzf,wzf}, sg[2]={-1.f,1.f};
    int vo[8]; float cf[8], dfx[8], dfy[8], dfz[8];
#pragma unroll
    for (int c=0;c<8;++c){
      int dx=c&1, dy=(c>>1)&1, dz=(c>>2)&1;
      vo[c]=(zi[dz]*RGRID+yi[dy])*RGRID+xi[dx];
      cf[c]=fx[dx]*fy[dy]*fz[dz];
      dfx[c]=sg[dx]*fy[dy]*fz[dz];
      dfy[c]=fx[dx]*sg[dy]*fz[dz];
      dfz[c]=fx[dx]*fy[dy]*sg[dz];
    }
    const float gsx=(pxu>0.f&&pxu<127.f)?0.5f*(float)RGRID*sclv:0.f;
    const float gsy=(pyu>0.f&&pyu<127.f)?0.5f*(float)RGRID*sclv:0.f;
    const float gsz=(pzu>0.f&&pzu<127.f)?0.5f*(float)RGRID*sclv:0.f;
    float sk0=0,sk1=0,sk2=0;
    float J00=0,J01=0,J02=0,J10=0,J11=0,J12=0,J20=0,J21=0,J22=0;
#pragma unroll
    for (int j=0;j<12;++j){
      const int ch=c0+j;
      float gv[8];
#pragma unroll
      for (int c=0;c<8;++c)
        gv[c]=TR? grid[(size_t)vo[c]*NCH+ch] : grid[(size_t)ch*NVOX+vo[c]];
      float w=0,rx=0,ry=0,rz=0;
#pragma unroll
      for (int c=0;c<8;++c){ w+=cf[c]*gv[c]; rx+=dfx[c]*gv[c]; ry+=dfy[c]*gv[c]; rz+=dfz[c]*gv[c]; }
      const float g0=rx*gsx, g1=ry*gsy, g2=rz*gsz;
      const float* Tn=Tmat+ch*12;
      const float t0=Tn[0],t1=Tn[1],t2=Tn[2],t3=Tn[3],t4=Tn[4],t5=Tn[5];
      const float t6=Tn[6],t7=Tn[7],t8=Tn[8],t9=Tn[9],t10=Tn[10],t11=Tn[11];
      const float s0=t0*x0+t1*x1+t2*x2+t3;
      const float s1=t4*x0+t5*x1+t6*x2+t7;
      const float s2=t8*x0+t9*x1+t10*x2+t11;
      sk0+=w*s0; sk1+=w*s1; sk2+=w*s2;
      J00+=g0*s0+w*t0; J01+=g1*s0+w*t1; J02+=g2*s0+w*t2;
      J10+=g0*s1+w*t4; J11+=g1*s1+w*t5; J12+=g2*s1+w*t6;
      J20+=g0*s2+w*t8; J21+=g1*s2+w*t9; J22+=g2*s2+w*t10;
    }
    // pair reduction: combine the two 12-channel halves (lane m <-> m+16)
    sk0+=__shfl_xor(sk0,16,32); sk1+=__shfl_xor(sk1,16,32); sk2+=__shfl_xor(sk2,16,32);
    J00+=__shfl_xor(J00,16,32); J01+=__shfl_xor(J01,16,32); J02+=__shfl_xor(J02,16,32);
    J10+=__shfl_xor(J10,16,32); J11+=__shfl_xor(J11,16,32); J12+=__shfl_xor(J12,16,32);
    J20+=__shfl_xor(J20,16,32); J21+=__shfl_xor(J21,16,32); J22+=__shfl_xor(J22,16,32);
    gx0=sk0-xd0; gx1=sk1-xd1; gx2=sk2-xd2;
    const float det=J00*(J11*J22-J12*J21)-J01*(J10*J22-J12*J20)+J02*(J10*J21-J11*J20);
    const float id=1.0f/det;
    Ji00=(J11*J22-J12*J21)*id; Ji01=(J02*J21-J01*J22)*id; Ji02=(J01*J12-J02*J11)*id;
    Ji10=(J12*J20-J10*J22)*id; Ji11=(J00*J22-J02*J20)*id; Ji12=(J02*J10-J00*J12)*id;
    Ji20=(J10*J21-J11*J20)*id; Ji21=(J01*J20-J00*J21)*id; Ji22=(J00*J11-J01*J10)*id;
  }

  // Broyden state
  float u0=-(Ji00*gx0+Ji01*gx1+Ji02*gx2);
  float u1=-(Ji10*gx0+Ji11*gx1+Ji12*gx2);
  float u2=-(Ji20*gx0+Ji21*gx1+Ji22*gx2);
  float xo0=x0, xo1=x1, xo2=x2;
  float gno=sqrtf(gx0*gx0+gx1*gx1+gx2*gx2);
  float dX0=0,dX1=0,dX2=0,dG0=0,dG1=0,dG2=0;
  int ids=1; unsigned alive=1u;

  for (int it=0; it<MAXSTEPS; ++it){
    if (!alive) break;                       // block-uniform early exit
    if (ids){ dX0=u0; dX1=u1; dX2=u2; x0+=dX0; x1+=dX1; x2+=dX2; }

    // ---- g(x): trilinear sample of this lane's 12 channels ----
    float vx=(x0-cen0)*sclv, vy=(x1-cen1)*sclv, vz=(x2-cen2)*sclv;
    float px=fminf(fmaxf(((vx+1.f)*(float)RGRID-1.f)*0.5f,0.f),127.f);
    float py=fminf(fmaxf(((vy+1.f)*(float)RGRID-1.f)*0.5f,0.f),127.f);
    float pz=fminf(fmaxf(((vz+1.f)*(float)RGRID-1.f)*0.5f,0.f),127.f);
    int ix0=(int)floorf(px), iy0=(int)floorf(py), iz0=(int)floorf(pz);
    float wxf=px-(float)ix0, wyf=py-(float)iy0, wzf=pz-(float)iz0;
    int xi[2]={ix0,min(ix0+1,RGRID-1)}, yi[2]={iy0,min(iy0+1,RGRID-1)}, zi[2]={iz0,min(iz0+1,RGRID-1)};
    float fx[2]={1.f-wxf,wxf}, fy[2]={1.f-wyf,wyf}, fz[2]={1.f-wzf,wzf};
    int vo[8]; float cf[8];
#pragma unroll
    for (int c=0;c<8;++c){
      int dx=c&1, dy=(c>>1)&1, dz=(c>>2)&1;
      vo[c]=(zi[dz]*RGRID+yi[dy])*RGRID+xi[dx];
      cf[c]=fx[dx]*fy[dy]*fz[dz];
    }
    float acc[12];
#pragma unroll
    for (int j=0;j<12;++j) acc[j]=0.f;
    if (TR){
#pragma unroll
      for (int c=0;c<8;++c){
        const float4* pv=(const float4*)(grid+(size_t)vo[c]*NCH+c0);
        const float4 q0=pv[0], q1=pv[1], q2=pv[2];
        const float w=cf[c];
        acc[0]+=w*q0.x; acc[1]+=w*q0.y; acc[2] +=w*q0.z; acc[3] +=w*q0.w;
        acc[4]+=w*q1.x; acc[5]+=w*q1.y; acc[6] +=w*q1.z; acc[7] +=w*q1.w;
        acc[8]+=w*q2.x; acc[9]+=w*q2.y; acc[10]+=w*q2.z; acc[11]+=w*q2.w;
      }
    } else {
#pragma unroll
      for (int j=0;j<12;++j){
        const float* gc=grid+(size_t)(c0+j)*NVOX;
        float a=0.f;
#pragma unroll
        for (int c=0;c<8;++c) a+=cf[c]*gc[vo[c]];
        acc[j]=a;
      }
    }

    // ---- stage weights, vote liveness, WMMA blend W @ T ----
    float4* wdst=(float4*)(ldsW + m*NCH + c0);
    wdst[0]=make_float4(acc[0],acc[1],acc[2],acc[3]);
    wdst[1]=make_float4(acc[4],acc[5],acc[6],acc[7]);
    wdst[2]=make_float4(acc[8],acc[9],acc[10],acc[11]);
    unsigned long long vb=__ballot(ids!=0);
    if (lane==0) s_vote[wv]=(unsigned)vb;
    __syncthreads();
    unsigned anext=0;
#pragma unroll
    for (int w2=0;w2<WPB;++w2) anext|=s_vote[w2];
    alive=anext;

    v8f Dacc = {};
#pragma unroll
    for (int kt=0;kt<6;++kt){
      const float2 aw=*(const float2*)(ldsW + m*NCH + 4*kt + 2*h);
      v2f Af; Af[0]=aw.x; Af[1]=aw.y;
      Dacc=__builtin_amdgcn_wmma_f32_16x16x4_f32(false, Af, false, Bf[kt],
                                                 (short)0, Dacc, false, false);
    }
    if (ncol<12){
#pragma unroll
      for (int r=0;r<8;++r) ldsD[(r+8*h)*12 + ncol]=Dacc[r];
    }
    __syncthreads();
    const float4 Ab0=*(const float4*)(ldsD + m*12 + 0);
    const float4 Ab1=*(const float4*)(ldsD + m*12 + 4);
    const float4 Ab2=*(const float4*)(ldsD + m*12 + 8);
    const float ng0=Ab0.x*x0+Ab0.y*x1+Ab0.z*x2+Ab0.w - xd0;
    const float ng1=Ab1.x*x0+Ab1.y*x1+Ab1.z*x2+Ab1.w - xd1;
    const float ng2=Ab2.x*x0+Ab2.y*x1+Ab2.z*x2+Ab2.w - xd2;

    // ---- Broyden update (redundant in lane pair) ----
    if (ids){ dG0=ng0-gx0; dG1=ng1-gx1; dG2=ng2-gx2; gx0=ng0; gx1=ng1; gx2=ng2; }
    const float gn=sqrtf(gx0*gx0+gx1*gx1+gx2*gx2);
    if (gn<gno){ gno=gn; xo0=x0; xo1=x1; xo2=x2; }
    ids=(gno>CVG_T)&&(gn<DVG_T);
    if (ids){
      const float v0=dX0*Ji00+dX1*Ji10+dX2*Ji20;
      const float v1=dX0*Ji01+dX1*Ji11+dX2*Ji21;
      const float v2=dX0*Ji02+dX1*Ji12+dX2*Ji22;
      const float a0=dX0-(Ji00*dG0+Ji01*dG1+Ji02*dG2);
      const float a1=dX1-(Ji10*dG0+Ji11*dG1+Ji12*dG2);
      const float a2=dX2-(Ji20*dG0+Ji21*dG1+Ji22*dG2);
      float b=v0*dG0+v1*dG1+v2*dG2;
      b+=(b>=0.f)?EPS_B:-EPS_B;
      const float ib=1.0f/b;
      Ji00+=a0*v0*ib; Ji01+=a0*v1*ib; Ji02+=a0*v2*ib;
      Ji10+=a1*v0*ib; Ji11+=a1*v1*ib; Ji12+=a1*v2*ib;
      Ji20+=a2*v0*ib; Ji21+=a2*v1*ib; Ji22+=a2*v2*ib;
    }
    u0=-(Ji00*gx0+Ji01*gx1+Ji02*gx2);
    u1=-(Ji10*gx0+Ji11*gx1+Ji12*gx2);
    u2=-(Ji20*gx0+Ji21*gx1+Ji22*gx2);
  }

  if (h==0){
    out[3*s+0]=xo0; out[3*s+1]=xo1; out[3*s+2]=xo2;
    out[3*NSOLVE + s]=(gno<CVG_T)?1.0f:0.0f;
  }
}

// ---------------------------------------------------------------------------
extern "C" void kernel_launch(void* const* d_in, const int* in_sizes, int n_in,
                              void* d_out, int out_size, void* d_ws, size_t ws_size,
                              hipStream_t stream) {
  (void)in_sizes; (void)n_in; (void)out_size;
  const float* xd   = (const float*)d_in[0];
  const float* tfs  = (const float*)d_in[1];
  const float* grid = (const float*)d_in[2];
  const float* ext  = (const float*)d_in[3];
  const float* cen  = (const float*)d_in[4];
  float* out = (float*)d_out;

  char* ws = (char*)d_ws;
  float* Tmat = (float*)(ws);            // 24*12 floats
  float* invT = (float*)(ws + 2048);     // 9*12 floats
  float* gT   = (float*)(ws + 4096);     // transposed grid (192 MB)
  const size_t need = 4096 + (size_t)NCH*NVOX*sizeof(float);

  deformer_setup_kernel<<<1, 32, 0, stream>>>(tfs, Tmat, invT);

  if (ws_size >= need) {
    grid_transpose_kernel<<<NVOX/256, 256, 0, stream>>>(grid, gT);
    broyden_solve_kernel<true><<<SOLVE_BLOCKS, THREADS, 0, stream>>>(
        xd, gT, Tmat, invT, ext, cen, out);
  } else {
    broyden_solve_kernel<false><<<SOLVE_BLOCKS, THREADS, 0, stream>>>(
        xd, grid, Tmat, invT, ext, cen, out);
  }
}